// StackAugmentedGRU_9981503996216
// MI455X (gfx1250) — compile-verified
//
#include <hip/hip_runtime.h>
#include <hip/hip_bf16.h>
#include <math.h>

// ---- problem dims ----
#define H   2048
#define SW  256
#define SD  200
#define OUT 128
#define KIH (H + SW)   // 2304
#define G3  (3 * H)    // 6144

// ---- ws layout (float offsets) ----
#define CTRL_OFF 0        // 3 floats: a_push, a_pop, a_noop
#define SIN_OFF  16       // 256 floats: tanh(Ws h + bs)
#define GIN_OFF  512      // 2304 floats: [emb_row(2048), stack_top(256)]
#define GI_OFF   4096     // 6144 floats: W_ih @ gin + b_ih
#define GH_OFF   12288    // 6144 floats: W_hh @ h + b_hh

typedef float v2f __attribute__((ext_vector_type(2)));
typedef float v8f __attribute__((ext_vector_type(8)));
typedef unsigned int v4u __attribute__((ext_vector_type(4)));
typedef int v8i __attribute__((ext_vector_type(8)));
typedef int v4i __attribute__((ext_vector_type(4)));

#if __has_builtin(__builtin_amdgcn_tensor_load_to_lds) && \
    __has_builtin(__builtin_amdgcn_s_wait_tensorcnt)
#define USE_TDM 1
#else
#define USE_TDM 0
#endif

// ---------------------------------------------------------------------------
// Kernel 1: ctrl softmax, stack-input projection, embedding gather.
//   blocks [0, SW)      : s_in row per block (2048-dot + tanh)
//   block  SW           : 3x2048 ctrl dots + softmax
//   blocks (SW, SW+8]   : copy emb[token] into gin[0:2048]
// ---------------------------------------------------------------------------
__global__ __launch_bounds__(256) void prep_kernel(
    const int* __restrict__ inp, const float* __restrict__ hidden,
    const float* __restrict__ emb, const float* __restrict__ Wc,
    const float* __restrict__ bc, const float* __restrict__ Ws,
    const float* __restrict__ bs, float* __restrict__ ws) {
  __shared__ float red[256];
  __shared__ float r0[256], r1[256], r2[256];
  const int tid = threadIdx.x;
  const int bx  = blockIdx.x;

  if (bx < SW) {
    const float* wrow = Ws + (size_t)bx * H;
    float acc = 0.f;
    for (int k = tid; k < H; k += 256) acc += wrow[k] * hidden[k];
    red[tid] = acc;
    __syncthreads();
    for (int s = 128; s > 0; s >>= 1) {
      if (tid < s) red[tid] += red[tid + s];
      __syncthreads();
    }
    if (tid == 0) ws[SIN_OFF + bx] = tanhf(red[0] + bs[bx]);
  } else if (bx == SW) {
    float a0 = 0.f, a1 = 0.f, a2 = 0.f;
    for (int k = tid; k < H; k += 256) {
      const float h = hidden[k];
      a0 += Wc[k] * h;
      a1 += Wc[H + k] * h;
      a2 += Wc[2 * H + k] * h;
    }
    r0[tid] = a0; r1[tid] = a1; r2[tid] = a2;
    __syncthreads();
    for (int s = 128; s > 0; s >>= 1) {
      if (tid < s) { r0[tid] += r0[tid + s]; r1[tid] += r1[tid + s]; r2[tid] += r2[tid + s]; }
      __syncthreads();
    }
    if (tid == 0) {
      const float l0 = r0[0] + bc[0], l1 = r1[0] + bc[1], l2 = r2[0] + bc[2];
      const float m  = fmaxf(l0, fmaxf(l1, l2));
      const float e0 = expf(l0 - m), e1 = expf(l1 - m), e2 = expf(l2 - m);
      const float inv = 1.0f / (e0 + e1 + e2);
      ws[CTRL_OFF + 0] = e0 * inv;  // a_push
      ws[CTRL_OFF + 1] = e1 * inv;  // a_pop
      ws[CTRL_OFF + 2] = e2 * inv;  // a_noop
    }
  } else {
    const int idx = (bx - SW - 1) * 256 + tid;  // 0..2047
    const int tok = inp[0];
    ws[GIN_OFF + idx] = emb[(size_t)tok * H + idx];
  }
}

// ---------------------------------------------------------------------------
// Kernel 2: differentiable stack push/pop/noop. grid = (SD), block = SW.
// Writes new_stack into d_out[128+H ...] and stack_top into gin[2048:2304].
// ---------------------------------------------------------------------------
__global__ __launch_bounds__(256) void stack_kernel(
    const float* __restrict__ stack, float* __restrict__ ws,
    float* __restrict__ out) {
  const int d = blockIdx.x;
  const int w = threadIdx.x;
  const float a_push = ws[CTRL_OFF + 0];
  const float a_pop  = ws[CTRL_OFF + 1];
  const float a_noop = ws[CTRL_OFF + 2];

  const float cur  = stack[d * SW + w];
  const float up   = (d == 0)      ? ws[SIN_OFF + w] : stack[(d - 1) * SW + w];
  const float down = (d < SD - 1)  ? stack[(d + 1) * SW + w] : 0.0f;
  const float v = a_noop * cur + a_push * up + a_pop * down;

  out[OUT + H + d * SW + w] = v;
  if (d == 0) ws[GIN_OFF + H + w] = v;  // stack_top -> tail of gin
}

// ---------------------------------------------------------------------------
// Kernel 3: memory-bound GEMV y = W x + b via V_WMMA_F32_16X16X4_F32.
// Block = 4 waves; each wave owns 16 output rows x one K-quarter, so the
// dependent WMMA accumulation chain is K/16 instructions and the grid exposes
// 4x the waves for HBM latency hiding. x is DMA'd into LDS once per block by
// the Tensor Data Mover (wave 0 issues tensor_load_to_lds, waits TENSORcnt,
// then the workgroup barrier publishes it). B fragment = x broadcast across
// all 16 N columns -> every column of D equals the 16 row-dots; lanes 0/16
// spill their C fragment to LDS and 16 threads do the 4-way partial sum.
// A-fragment per lane: row = lane&15, cols = k + 2*(lane>>4) (global b64).
// K/4 is a multiple of 32 for all call sites, so EXEC is all-1s throughout.
// ---------------------------------------------------------------------------
__global__ __launch_bounds__(128) void wmma_gemv_f32(
    const float* __restrict__ W, int ldw, int K,
    const float* __restrict__ x, const float* __restrict__ bias,
    float* __restrict__ y) {
  __shared__ float xs[KIH];
  __shared__ float part[4][16];
  const int lane = threadIdx.x & 31;
  const int wave = threadIdx.x >> 5;
  const int tile = blockIdx.x;   // 16 rows per tile

#if USE_TDM
  if (wave == 0) {
    // Tensor DMA descriptor: 1-row 2D tile, 4-byte elements, length K.
    const unsigned long long ga = (unsigned long long)(const void*)x;
    const unsigned int lds = (unsigned int)(unsigned long long)(const void*)xs;
    v4u g0;
    g0[0] = 1u;                                             // count=1, user D#
    g0[1] = lds;                                            // lds_addr
    g0[2] = (unsigned int)ga;                               // global_addr lo
    g0[3] = (unsigned int)((ga >> 32) & 0x01FFFFFFu) | (2u << 30);  // hi + type=2
    v8i g1;
    g1[0] = (int)(2u << 16);                 // data_size=4B, no multicast
    g1[1] = (int)((unsigned)(K & 0xFFFF) << 16);   // tensor_dim0[15:0]
    g1[2] = (int)(((unsigned)K >> 16) & 0xFFFFu) | (1 << 16);  // dim0 hi | tensor_dim1=1
    g1[3] = (int)((unsigned)(K & 0xFFFF) << 16);   // tile_dim0 = K
    g1[4] = 1;                               // tile_dim1=1, tile_dim2=0
    g1[5] = K;                               // tensor_dim0_stride lo
    g1[6] = 0;                               // stride hi, dim1_stride lo
    g1[7] = 0;
    v4i z4 = {0, 0, 0, 0};
    v8i z8 = {0, 0, 0, 0, 0, 0, 0, 0};
    __builtin_amdgcn_tensor_load_to_lds(g0, g1, z4, z4, z8, 0);
    __builtin_amdgcn_s_wait_tensorcnt(0);
  }
  __syncthreads();
#else
  for (int k = threadIdx.x; k < K; k += 128) xs[k] = x[k];
  __syncthreads();
#endif

  const int kq     = K >> 2;                 // per-wave K slice
  const int k0     = wave * kq;
  const int row    = tile * 16 + (lane & 15);
  const int colsub = 2 * (lane >> 4);
  const float* wp = W + (size_t)row * ldw + k0 + colsub;
  const float* xp = xs + k0 + colsub;

  v8f c = {0.f, 0.f, 0.f, 0.f, 0.f, 0.f, 0.f, 0.f};

#if __has_builtin(__builtin_amdgcn_wmma_f32_16x16x4_f32)
#pragma unroll 8
  for (int k = 0; k < kq; k += 4) {
    v2f a = *(const v2f*)(wp + k);   // 16x4 f32 A tile (256B/wave/wmma)
    v2f b = *(const v2f*)(xp + k);   // x pair, broadcast over N
    c = __builtin_amdgcn_wmma_f32_16x16x4_f32(
        /*neg_a=*/false, a, /*neg_b=*/false, b,
        /*c_mod=*/(short)0, c, /*reuse_a=*/false, /*reuse_b=*/false);
  }
  if ((lane & 15) == 0) {
    const int b8 = (lane >> 4) << 3;
#pragma unroll
    for (int r = 0; r < 8; ++r) part[wave][b8 + r] = c[r];
  }
#else
  float acc = 0.f;
#pragma unroll 8
  for (int k = 0; k < kq; k += 4) {
    v2f a = *(const v2f*)(wp + k);
    v2f b = *(const v2f*)(xp + k);
    acc += a[0] * b[0] + a[1] * b[1];
  }
  acc += __shfl_xor(acc, 16);
  if (lane < 16) part[wave][lane] = acc;
#endif

  __syncthreads();
  if (threadIdx.x < 16) {
    const int r = threadIdx.x;
    y[tile * 16 + r] = part[0][r] + part[1][r] + part[2][r] + part[3][r] +
                       bias[tile * 16 + r];
  }
}

// ---------------------------------------------------------------------------
// Kernel 4: GRU gates (PyTorch order r,z,n). grid = 8, block = 256.
// ---------------------------------------------------------------------------
__global__ __launch_bounds__(256) void gates_kernel(
    const float* __restrict__ ws, const float* __restrict__ hidden,
    float* __restrict__ out) {
  const int i = blockIdx.x * 256 + threadIdx.x;  // 0..2047
  const float* gi = ws + GI_OFF;
  const float* gh = ws + GH_OFF;
  const float r = 1.f / (1.f + expf(-(gi[i] + gh[i])));
  const float z = 1.f / (1.f + expf(-(gi[H + i] + gh[H + i])));
  const float n = tanhf(gi[2 * H + i] + r * gh[2 * H + i]);
  out[OUT + i] = (1.f - z) * n + z * hidden[i];  // h_new
}

// ---------------------------------------------------------------------------
extern "C" void kernel_launch(void* const* d_in, const int* in_sizes, int n_in,
                              void* d_out, int out_size, void* d_ws, size_t ws_size,
                              hipStream_t stream) {
  const int*   inp    = (const int*)  d_in[0];
  const float* hidden = (const float*)d_in[1];
  const float* stack  = (const float*)d_in[2];
  const float* emb    = (const float*)d_in[3];
  const float* Wc     = (const float*)d_in[4];
  const float* bc     = (const float*)d_in[5];
  const float* Ws     = (const float*)d_in[6];
  const float* bs     = (const float*)d_in[7];
  const float* W_ih   = (const float*)d_in[8];
  const float* b_ih   = (const float*)d_in[9];
  const float* W_hh   = (const float*)d_in[10];
  const float* b_hh   = (const float*)d_in[11];
  const float* Wd     = (const float*)d_in[12];
  const float* bd     = (const float*)d_in[13];
  float* out = (float*)d_out;
  float* ws  = (float*)d_ws;

  // 1) ctrl softmax + s_in + emb gather
  prep_kernel<<<SW + 1 + 8, 256, 0, stream>>>(inp, hidden, emb, Wc, bc, Ws, bs, ws);
  // 2) gh = W_hh h + b_hh  (depends only on hidden; 50 MB stream, WMMA)
  wmma_gemv_f32<<<G3 / 16, 128, 0, stream>>>(W_hh, H, H, hidden, b_hh, ws + GH_OFF);
  // 3) stack blend -> d_out, stack_top -> gin tail
  stack_kernel<<<SD, SW, 0, stream>>>(stack, ws, out);
  // 4) gi = W_ih gin + b_ih  (57 MB stream, WMMA)
  wmma_gemv_f32<<<G3 / 16, 128, 0, stream>>>(W_ih, KIH, KIH, ws + GIN_OFF, b_ih, ws + GI_OFF);
  // 5) gates -> h_new (d_out[128:128+H])
  gates_kernel<<<H / 256, 256, 0, stream>>>(ws, hidden, out);
  // 6) out = Wd h_new + bd (d_out[0:128])
  wmma_gemv_f32<<<OUT / 16, 128, 0, stream>>>(Wd, H, H, out + OUT, bd, out);
}